// PGDStep_8383776161795
// MI455X (gfx1250) — compile-verified
//
#include <hip/hip_runtime.h>
#include <cstdint>

// ---------------------------------------------------------------------------
// PGD step + knapsack water-filling projection, one block per row.
// B=16384 rows, N=4096 cols, f32.
//   y      = max(d + 0.05*sqrt(r*g)/sqrt(max(d,1e-12)), 0)
//   cost   = sum(y*c);  if cost > Q: 25x bisection on lambda,
//   excess = sum(max(y - lam*c, 0)*c) - Q;  out = max(y - lam_hi*c, 0)
// Row data lives in VGPRs for all 25 iterations (zero re-read traffic).
// c and g (shared column vectors) are prefetched via CDNA5 async
// global->LDS DMA overlapping the cold HBM loads of r and d.
// ---------------------------------------------------------------------------

#define TPB   128
#define NW    (TPB / 32)               // 4 wave32s per block
#define NCOLS 4096
#define CPT   (NCOLS / (TPB * 4))      // 8 float4 chunks per thread
#define EPT   (CPT * 4)                // 32 elements per thread

static constexpr float QBUD  = 100.0f;
static constexpr float LRATE = 0.05f;
static constexpr float EPSV  = 1e-12f;
static constexpr int   NITER = 25;

// CDNA5 async global->LDS copy (VGLOBAL, tracked by ASYNCcnt).
__device__ __forceinline__ void async_b128_to_lds(const float* gsrc,
                                                  unsigned lds_byte_addr) {
  unsigned long long ga = (unsigned long long)(uintptr_t)gsrc;
  asm volatile("global_load_async_to_lds_b128 %0, %1, off"
               :: "v"(lds_byte_addr), "v"(ga)
               : "memory");
}

__device__ __forceinline__ void wait_async_zero() {
  asm volatile("s_wait_asynccnt 0x0" ::: "memory");
}

__device__ __forceinline__ float wave_sum(float v) {
#pragma unroll
  for (int off = 16; off > 0; off >>= 1) v += __shfl_xor(v, off, 32);
  return v;
}
__device__ __forceinline__ float wave_max(float v) {
#pragma unroll
  for (int off = 16; off > 0; off >>= 1) v = fmaxf(v, __shfl_xor(v, off, 32));
  return v;
}

// Deterministic block reduction: every thread ends with the bitwise-identical
// result, so bisection branches are uniform across the block.
__device__ __forceinline__ float block_sum(float v, float* red, int lane, int wid) {
  v = wave_sum(v);
  __syncthreads();                 // protect red[] from previous round's readers
  if (lane == 0) red[wid] = v;
  __syncthreads();
  float s = red[0];
#pragma unroll
  for (int w = 1; w < NW; ++w) s += red[w];
  return s;
}
__device__ __forceinline__ float block_max(float v, float* red, int lane, int wid) {
  v = wave_max(v);
  __syncthreads();
  if (lane == 0) red[wid] = v;
  __syncthreads();
  float s = red[0];
#pragma unroll
  for (int w = 1; w < NW; ++w) s = fmaxf(s, red[w]);
  return s;
}

__global__ void __launch_bounds__(TPB)
pgd_knapsack_kernel(const float* __restrict__ r, const float* __restrict__ d,
                    const float* __restrict__ g, const float* __restrict__ c_cost,
                    float* __restrict__ out) {
  __shared__ float lds_c[NCOLS];   // 16 KB
  __shared__ float lds_g[NCOLS];   // 16 KB
  __shared__ float red[NW];

  const int t    = threadIdx.x;
  const int lane = t & 31;
  const int wid  = t >> 5;
  const long long row = blockIdx.x;

  // Escape LDS base addresses (low 32 bits of the generic pointer = LDS byte
  // offset per the flat-aperture mapping) and kick off async DMA of c and g.
  const unsigned c_base = (unsigned)(uintptr_t)&lds_c[0];
  const unsigned g_base = (unsigned)(uintptr_t)&lds_g[0];
#pragma unroll
  for (int k = 0; k < CPT; ++k) {
    const int chunk = t + k * TPB;                    // float4 index
    async_b128_to_lds(c_cost + 4 * chunk, c_base + 16u * (unsigned)chunk);
    async_b128_to_lds(g      + 4 * chunk, g_base + 16u * (unsigned)chunk);
  }

  // Cold HBM loads of this row while the async copies fly (L2-hot vectors).
  const float4* r4 = (const float4*)(r + row * (long long)NCOLS);
  const float4* d4 = (const float4*)(d + row * (long long)NCOLS);
  float4 rv[CPT], dv[CPT];
#pragma unroll
  for (int k = 0; k < CPT; ++k) {
    rv[k] = r4[t + k * TPB];
    dv[k] = d4[t + k * TPB];
  }

  wait_async_zero();     // this wave's async copies landed in LDS
  __syncthreads();       // all waves' copies visible block-wide

  // PGD step into registers: y[e], cw[e] stay live for all 25 iterations.
  float y[EPT], cw[EPT];
  const float4* c4 = (const float4*)lds_c;
  const float4* g4 = (const float4*)lds_g;
#pragma unroll
  for (int k = 0; k < CPT; ++k) {
    const float4 cv = c4[t + k * TPB];
    const float4 gv = g4[t + k * TPB];
    const float rr[4] = {rv[k].x, rv[k].y, rv[k].z, rv[k].w};
    const float dd[4] = {dv[k].x, dv[k].y, dv[k].z, dv[k].w};
    const float gg[4] = {gv.x, gv.y, gv.z, gv.w};
    const float cc[4] = {cv.x, cv.y, cv.z, cv.w};
#pragma unroll
    for (int i = 0; i < 4; ++i) {
      const int e = k * 4 + i;
      const float u  = rr[i] * gg[i];                       // util = r*g
      const float m  = fmaxf(dd[i], EPSV);
      const float dn = dd[i] + LRATE * (sqrtf(u) / sqrtf(m)); // d - LR*grad
      y[e]  = fmaxf(dn, 0.0f);
      cw[e] = cc[i];
    }
  }

  // cost = sum(y*c) and lam_hi0 = max(y/c), 4-way accumulators.
  float a0 = 0.f, a1 = 0.f, a2 = 0.f, a3 = 0.f, mx = 0.f;
#pragma unroll
  for (int e = 0; e < EPT; e += 4) {
    a0 = fmaf(y[e + 0], cw[e + 0], a0);
    a1 = fmaf(y[e + 1], cw[e + 1], a1);
    a2 = fmaf(y[e + 2], cw[e + 2], a2);
    a3 = fmaf(y[e + 3], cw[e + 3], a3);
    mx = fmaxf(mx, y[e + 0] / cw[e + 0]);
    mx = fmaxf(mx, y[e + 1] / cw[e + 1]);
    mx = fmaxf(mx, y[e + 2] / cw[e + 2]);
    mx = fmaxf(mx, y[e + 3] / cw[e + 3]);
  }
  const float cost = block_sum((a0 + a1) + (a2 + a3), red, lane, wid);
  float hi = block_max(mx, red, lane, wid);
  float lo = 0.0f;
  const bool viol = cost > QBUD;   // block-uniform (deterministic reduction)

  if (viol) {
    for (int it = 0; it < NITER; ++it) {
      const float lam = 0.5f * (lo + hi);
      // max(y - lam*c, 0)*c  ==  3 VALU ops/element: fma, max, fma
      float b0 = 0.f, b1 = 0.f, b2 = 0.f, b3 = 0.f;
#pragma unroll
      for (int e = 0; e < EPT; e += 4) {
        const float t0 = fmaxf(fmaf(-lam, cw[e + 0], y[e + 0]), 0.0f);
        const float t1 = fmaxf(fmaf(-lam, cw[e + 1], y[e + 1]), 0.0f);
        const float t2 = fmaxf(fmaf(-lam, cw[e + 2], y[e + 2]), 0.0f);
        const float t3 = fmaxf(fmaf(-lam, cw[e + 3], y[e + 3]), 0.0f);
        b0 = fmaf(t0, cw[e + 0], b0);
        b1 = fmaf(t1, cw[e + 1], b1);
        b2 = fmaf(t2, cw[e + 2], b2);
        b3 = fmaf(t3, cw[e + 3], b3);
      }
      const float excess =
          block_sum((b0 + b1) + (b2 + b3), red, lane, wid) - QBUD;
      if (excess > 0.0f) lo = lam; else hi = lam;   // identical on all threads
    }
  }

  // Final projection with lam_hi; rows under budget pass y through.
  float4* o4 = (float4*)(out + row * (long long)NCOLS);
#pragma unroll
  for (int k = 0; k < CPT; ++k) {
    float v[4];
#pragma unroll
    for (int i = 0; i < 4; ++i) {
      const int e = k * 4 + i;
      v[i] = viol ? fmaxf(fmaf(-hi, cw[e], y[e]), 0.0f) : y[e];
    }
    float4 ov;
    ov.x = v[0]; ov.y = v[1]; ov.z = v[2]; ov.w = v[3];
    o4[t + k * TPB] = ov;
  }
}

extern "C" void kernel_launch(void* const* d_in, const int* in_sizes, int n_in,
                              void* d_out, int out_size, void* d_ws, size_t ws_size,
                              hipStream_t stream) {
  (void)n_in; (void)d_ws; (void)ws_size; (void)out_size;
  const float* r  = (const float*)d_in[0];
  const float* d  = (const float*)d_in[1];
  const float* g  = (const float*)d_in[2];
  const float* cc = (const float*)d_in[3];
  float* out = (float*)d_out;

  const int n    = in_sizes[2] > 0 ? in_sizes[2] : NCOLS;   // 4096
  const int rows = in_sizes[0] / n;                         // 16384

  pgd_knapsack_kernel<<<rows, TPB, 0, stream>>>(r, d, g, cc, out);
}